// Decoder_41540923687243
// MI455X (gfx1250) — compile-verified
//
#include <hip/hip_runtime.h>
#include <math.h>

// Problem constants (match reference)
#define BB 512
#define NN 1000
#define DD 128
#define HH 16
#define DK 8
#define NT 1024          // padded N for the compat LDS buffer
#define NTILES 63        // ceil(1000/16)

typedef float v2f __attribute__((ext_vector_type(2)));
typedef float v8f __attribute__((ext_vector_type(8)));

__global__ void __launch_bounds__(256)
decoder_fused_kernel(const float* __restrict__ emb,     // [B,N,D]
                     const float* __restrict__ Wnode,   // [3D,D]
                     const float* __restrict__ Wfixed,  // [D,D]
                     const float* __restrict__ Wstep,   // [D,2D]
                     const float* __restrict__ Wout,    // [D,D]
                     const int*   __restrict__ first_a, // [B]
                     const int*   __restrict__ curnode, // [B]
                     const unsigned char* __restrict__ mask, // [B,1,N] bool
                     float*       __restrict__ out)     // [B,1,N]
{
    extern __shared__ float smem[];                 // 16*1024 f32 = 64KB: compat / scratch / logits
    __shared__ float qh[HH][DD];                    // qh_proj  (8KB)
    __shared__ float Ahd[HH][DD];                   // attn @ emb (8KB)
    __shared__ float ge[DD];
    __shared__ float stepctx[2 * DD];
    __shared__ float query[DD];
    __shared__ float headsg[DD];
    __shared__ float glimpse[DD];
    __shared__ float uvec[DD];
    __shared__ float red[16];
    __shared__ unsigned char mskS[NN];              // mask staged in LDS (1000 B)

    const int b    = blockIdx.x;
    const int tid  = threadIdx.x;
    const int lane = tid & 31;
    const int wid  = tid >> 5;                      // 8 waves
    const int lrow  = lane & 15;                    // 0..15
    const int lhalf = lane >> 4;                    // 0 or 1

    const float* __restrict__ E  = emb  + (size_t)b * NN * DD;
    const unsigned char* __restrict__ mk = mask + (size_t)b * NN;
    float (*compat)[NT] = reinterpret_cast<float (*)[NT]>(smem);

    // ---------------- Phase 1: graph_embed = mean_n emb[b,n,:] ----------------
    {
        const int col4 = tid & 31;                  // float4 column (0..31)
        const int rg   = tid >> 5;                  // row group (0..7)
        float4 acc = make_float4(0.f, 0.f, 0.f, 0.f);
        for (int n = rg; n < NN; n += 8) {
            const float4 v = ((const float4*)(E + (size_t)n * DD))[col4];
            acc.x += v.x; acc.y += v.y; acc.z += v.z; acc.w += v.w;
        }
        float* sc = smem + rg * DD + col4 * 4;      // reuse compat buffer as scratch
        sc[0] = acc.x; sc[1] = acc.y; sc[2] = acc.z; sc[3] = acc.w;
    }
    // stage the feasibility mask into LDS while phase-1 partials land
    for (int j = tid; j < NN; j += 256) mskS[j] = mk[j];
    __syncthreads();
    if (tid < DD) {
        float s = 0.f;
        #pragma unroll
        for (int g = 0; g < 8; ++g) s += smem[g * DD + tid];
        ge[tid] = s * (1.0f / (float)NN);
    }
    // gather step context rows
    {
        const int i0 = first_a[b];
        const int i1 = curnode[b];
        for (int j = tid; j < DD; j += 256) {
            stepctx[j]      = E[(size_t)i0 * DD + j];
            stepctx[DD + j] = E[(size_t)i1 * DD + j];
        }
    }
    __syncthreads();

    // ---------------- Phase 2: query and qh_proj ----------------
    if (tid < DD) {
        float s = 0.f;
        const float* wf = Wfixed + (size_t)tid * DD;
        #pragma unroll 4
        for (int d = 0; d < DD; ++d) s += wf[d] * ge[d];
        const float* ws = Wstep + (size_t)tid * 2 * DD;
        #pragma unroll 4
        for (int j = 0; j < 2 * DD; ++j) s += ws[j] * stepctx[j];
        query[tid] = s;
    }
    __syncthreads();
    // qh_proj[h][d] = sum_k Q[h,k] * W_gK[h*8+k, d]
    for (int e = tid; e < HH * DD; e += 256) {
        const int h = e >> 7;
        const int d = e & 127;
        float s = 0.f;
        #pragma unroll
        for (int k = 0; k < DK; ++k)
            s += query[h * DK + k] * Wnode[(size_t)(h * DK + k) * DD + d];
        qh[h][d] = s;
    }
    __syncthreads();

    // ---------------- Phase 3: compat[h,n] = qh_proj[h,:]·emb[n,:] / sqrt(dk) (WMMA) ----
    // WMMA roles: A = emb tile (M=n rows, K=d), B = qh_proj (K=d, N=h). D: M=n, N=h.
    {
        const float scale = 0.35355339059327378f;   // 1/sqrt(8)
        for (int t = wid; t < NTILES; t += 8) {
            const int n0 = t * 16;
            int arow = n0 + lrow;                   // A-matrix row this lane supplies
            if (arow > NN - 1) arow = NN - 1;       // clamp tail (results ignored)
            const float* Ar = E + (size_t)arow * DD;
            v8f acc = {};
            #pragma unroll 4
            for (int k0 = 0; k0 < DD; k0 += 4) {
                const int kk = k0 + 2 * lhalf;
                v2f a = *(const v2f*)(Ar + kk);                 // A: K=kk, kk+1
                v2f bf = *(const v2f*)(&qh[lrow][kk]);          // B: N=h=lrow, K=kk,kk+1
                acc = __builtin_amdgcn_wmma_f32_16x16x4_f32(
                        false, a, false, bf, (short)0, acc, false, false);
            }
            // D layout: VGPR r -> (M = r + 8*lhalf, N = lrow); row of compat is contiguous in n
            float* dst = &compat[lrow][n0 + 8 * lhalf];
            #pragma unroll
            for (int r = 0; r < 8; ++r) dst[r] = acc[r] * scale;
        }
    }
    __syncthreads();

    // ---------------- Phase 4: masked softmax per head -> attn stored back in compat ----
    {
        for (int i = 0; i < 2; ++i) {
            const int h = wid * 2 + i;
            float m = -INFINITY;
            for (int n = lane; n < NN; n += 32) {
                float c = mskS[n] ? -INFINITY : compat[h][n];
                compat[h][n] = c;
                m = fmaxf(m, c);
            }
            #pragma unroll
            for (int off = 16; off >= 1; off >>= 1)
                m = fmaxf(m, __shfl_xor(m, off, 32));
            float l = 0.f;
            for (int n = lane; n < NN; n += 32) {
                const float e = expf(compat[h][n] - m);
                compat[h][n] = e;                   // store unnormalized weight
                l += e;
            }
            #pragma unroll
            for (int off = 16; off >= 1; off >>= 1)
                l += __shfl_xor(l, off, 32);
            const float inv_l = 1.0f / l;
            for (int n = lane; n < NN; n += 32)
                compat[h][n] *= inv_l;
        }
    }
    __syncthreads();

    // ---------------- Phase 5: A[h,d] = sum_n attn[h,n] * emb[n,d] (WMMA) ----------------
    // Roles: A-matrix = attn (M=h, K=n), B = emb (K=n, N=d). D: M=h, N=d. K=1000 = 250*4.
    {
        const int d0 = wid * 16;                    // one 16-wide d tile per wave
        const int dcol = d0 + lrow;
        v8f acc = {};
        #pragma unroll 4
        for (int k0 = 0; k0 < NN; k0 += 4) {
            const int kk = k0 + 2 * lhalf;
            v2f a = *(const v2f*)(&compat[lrow][kk]);           // attn[h=lrow][kk,kk+1]
            v2f bf;
            bf.x = E[(size_t)kk * DD + dcol];                   // emb[kk,   d]
            bf.y = E[(size_t)(kk + 1) * DD + dcol];             // emb[kk+1, d]
            acc = __builtin_amdgcn_wmma_f32_16x16x4_f32(
                    false, a, false, bf, (short)0, acc, false, false);
        }
        #pragma unroll
        for (int r = 0; r < 8; ++r) Ahd[r + 8 * lhalf][dcol] = acc[r];
    }
    __syncthreads();

    // ---------------- Phase 6: heads, glimpse, u = W_lK^T @ glimpse ----------------
    if (tid < HH * DK) {                            // 128 threads: heads[h*8+k]
        const int h = tid >> 3;
        const float* wv = Wnode + (size_t)(DD + tid) * DD;
        float s = 0.f;
        #pragma unroll 4
        for (int d = 0; d < DD; ++d) s += Ahd[h][d] * wv[d];
        headsg[tid] = s;
    }
    __syncthreads();
    if (tid < DD) {
        float s = 0.f;
        const float* wo = Wout + (size_t)tid * DD;
        #pragma unroll 4
        for (int j = 0; j < DD; ++j) s += wo[j] * headsg[j];
        glimpse[tid] = s;
    }
    __syncthreads();
    if (tid < DD) {
        float s = 0.f;
        #pragma unroll 4
        for (int e = 0; e < DD; ++e)
            s += glimpse[e] * Wnode[(size_t)(2 * DD + e) * DD + tid];
        uvec[tid] = s;
    }
    __syncthreads();

    // ---------------- Phase 7: logits + log_softmax ----------------
    float* lg = smem;                               // reuse compat buffer for logits[N]
    {
        const float inv_sqrt_d = 0.08838834764831845f;  // 1/sqrt(128)
        for (int n = tid; n < NN; n += 256) {
            const float4* er = (const float4*)(E + (size_t)n * DD);
            float s = 0.f;
            #pragma unroll 8
            for (int j = 0; j < DD / 4; ++j) {
                const float4 v = er[j];
                s += v.x * uvec[4 * j] + v.y * uvec[4 * j + 1]
                   + v.z * uvec[4 * j + 2] + v.w * uvec[4 * j + 3];
            }
            const float t = tanhf(s * inv_sqrt_d) * 10.0f;
            lg[n] = mskS[n] ? -INFINITY : t;
        }
    }
    __syncthreads();
    // block-wide max
    float m = -INFINITY;
    for (int n = tid; n < NN; n += 256) m = fmaxf(m, lg[n]);
    #pragma unroll
    for (int off = 16; off >= 1; off >>= 1) m = fmaxf(m, __shfl_xor(m, off, 32));
    if (lane == 0) red[wid] = m;
    __syncthreads();
    float M = -INFINITY;
    #pragma unroll
    for (int w = 0; w < 8; ++w) M = fmaxf(M, red[w]);
    __syncthreads();
    // block-wide sum of exp
    float l = 0.f;
    for (int n = tid; n < NN; n += 256) l += expf(lg[n] - M);
    #pragma unroll
    for (int off = 16; off >= 1; off >>= 1) l += __shfl_xor(l, off, 32);
    if (lane == 0) red[wid] = l;
    __syncthreads();
    float L = 0.f;
    #pragma unroll
    for (int w = 0; w < 8; ++w) L += red[w];
    const float logZ = M + logf(L);
    float* ob = out + (size_t)b * NN;
    for (int n = tid; n < NN; n += 256) ob[n] = lg[n] - logZ;
}

extern "C" void kernel_launch(void* const* d_in, const int* in_sizes, int n_in,
                              void* d_out, int out_size, void* d_ws, size_t ws_size,
                              hipStream_t stream) {
    const float* emb    = (const float*)d_in[0];   // [512,1000,128] f32
    const float* Wnode  = (const float*)d_in[1];   // [384,128] f32
    const float* Wfixed = (const float*)d_in[2];   // [128,128] f32
    const float* Wstep  = (const float*)d_in[3];   // [128,256] f32
    const float* Wout   = (const float*)d_in[4];   // [128,128] f32
    const int*   fa     = (const int*)d_in[5];     // [512,1] i32
    const int*   cn     = (const int*)d_in[6];     // [512,1] i32
    const unsigned char* mask = (const unsigned char*)d_in[7]; // [512,1,1000] bool
    float* out = (float*)d_out;                    // [512,1,1000] f32

    const size_t dyn_lds = (size_t)HH * NT * sizeof(float);    // 64 KB compat buffer
    decoder_fused_kernel<<<BB, 256, dyn_lds, stream>>>(
        emb, Wnode, Wfixed, Wstep, Wout, fa, cn, mask, out);
}